// SparsifiedLinearFP8_72292889526732
// MI455X (gfx1250) — compile-verified
//
#include <hip/hip_runtime.h>
#include <hip/hip_fp16.h>

typedef float    v2f __attribute__((ext_vector_type(2)));
typedef float    v8f __attribute__((ext_vector_type(8)));
typedef int      v2i __attribute__((ext_vector_type(2)));
typedef _Float16 v2h __attribute__((ext_vector_type(2)));

#define BATCH 16
#define CHUNK 256              // nnz per staged buffer: cols 1KB (64x16B), vals 512B (32x16B)
#define WAVES_PER_BLOCK 8
#define THREADS (WAVES_PER_BLOCK * 32)

// ---- CDNA5 async global->LDS copy (ASYNCcnt-tracked), 16B per lane ----
__device__ __forceinline__ void async_copy_b128(void* lds, const void* gaddr) {
    // low 32 bits of a generic pointer to __shared__ == LDS byte offset (flat-LDS aperture)
    unsigned lds_off = (unsigned)(unsigned long long)lds;
    asm volatile("global_load_async_to_lds_b128 %0, %1, off"
                 :: "v"(lds_off), "v"(gaddr) : "memory");
}
__device__ __forceinline__ void wait_async_le3() {
#if __has_builtin(__builtin_amdgcn_s_wait_asynccnt)
    __builtin_amdgcn_s_wait_asynccnt(3);
#else
    asm volatile("s_wait_asynccnt 3" ::: "memory");
#endif
    asm volatile("" ::: "memory");
}
__device__ __forceinline__ void wait_async_0() {
#if __has_builtin(__builtin_amdgcn_s_wait_asynccnt)
    __builtin_amdgcn_s_wait_asynccnt(0);
#else
    asm volatile("s_wait_asynccnt 0" ::: "memory");
#endif
    asm volatile("" ::: "memory");
}

// ---- x[16][N] f16  ->  xT[N][16] f32 ----
__global__ void __launch_bounds__(256)
transpose_x_kernel(const _Float16* __restrict__ x, float* __restrict__ xT, int N) {
    int tid = blockIdx.x * blockDim.x + threadIdx.x;
    int total = N * BATCH;
    if (tid >= total) return;
    int b = tid / N;            // batch row (coalesced read of x)
    int n = tid - b * N;
    xT[(size_t)n * BATCH + b] = (float)x[tid];
}

// ---- CSR SpMM: one wave per output row; lanes0-15=batch.
//      Each half-wave owns a consecutive nnz PAIR per step (4 nnz / step):
//      cols pair = 1 ds_load_b64, vals pair = 1 ds_load_b32, f16 converts
//      folded into v_fma_mix op_sel lo/hi. ----
template<bool OUT_F16>
__global__ void __launch_bounds__(THREADS)
spmm_kernel(const int* __restrict__ row_off,
            const int* __restrict__ cols,
            const _Float16* __restrict__ vals,
            const float* __restrict__ xT,   // [n_cols][BATCH] f32, L2-resident
            void* __restrict__ outp,
            int n_rows) {
    __shared__ __align__(16) int      s_cols[WAVES_PER_BLOCK][2][CHUNK];
    __shared__ __align__(16) _Float16 s_vals[WAVES_PER_BLOCK][2][CHUNK];

    const int lane = threadIdx.x & 31;
    const int w    = threadIdx.x >> 5;
    const int row  = blockIdx.x * WAVES_PER_BLOCK + w;
    if (row >= n_rows) return;

    const int start = row_off[row];
    const int end   = row_off[row + 1];
    const int nnz   = end - start;
    const int half  = lane >> 4;    // which nnz pair of the quad
    const int b     = lane & 15;    // batch index
    const int nchunks = (nnz + CHUNK - 1) / CHUNK;

    // Always exactly 3 async instructions (double-buffer accounting stays exact);
    // clamped segment indices instead of lane guards -> no EXEC divergence, and
    // duplicate 16B copies of identical data are benign.
    auto issue = [&](int ci, int buf) {
        int c     = start + ci * CHUNK;
        int r     = min(CHUNK, end - c);
        int csegs = (r + 3) >> 2;        // 16B segments of cols (<= 64)
        int vsegs = (r + 7) >> 3;        // 16B segments of vals (<= 32)
        int s1 = min(lane,      csegs - 1);
        int s2 = min(lane + 32, csegs - 1);
        int sv = min(lane,      vsegs - 1);
        async_copy_b128(&s_cols[w][buf][s1 * 4], cols + c + s1 * 4);
        async_copy_b128(&s_cols[w][buf][s2 * 4], cols + c + s2 * 4);
        async_copy_b128(&s_vals[w][buf][sv * 8], vals + c + sv * 8);
    };

    if (nchunks > 0) issue(0, 0);

    float acc[4] = {0.0f, 0.0f, 0.0f, 0.0f};

    // one quad-step: half-wave h consumes nnz {jj+2h, jj+2h+1}
    auto quad = [&](const int* cb, const _Float16* vb, int jj, int u) {
        const int idx = jj + 2 * half;                 // even -> 8B/4B aligned
        const v2i cp  = *(const v2i*)(cb + idx);       // ds_load_b64
        const v2h vp  = *(const v2h*)(vb + idx);       // ds_load_b32
        const float xv0 = xT[(size_t)cp.x * BATCH + b];
        const float xv1 = xT[(size_t)cp.y * BATCH + b];
        acc[u + 0] = fmaf((float)vp.x, xv0, acc[u + 0]);
        acc[u + 1] = fmaf((float)vp.y, xv1, acc[u + 1]);
    };

    for (int ci = 0; ci < nchunks; ++ci) {
        const int buf  = ci & 1;
        const int c    = start + ci * CHUNK;
        const int r    = min(CHUNK, end - c);
        const bool pref = (ci + 1 < nchunks);
        if (pref) { issue(ci + 1, buf ^ 1); wait_async_le3(); }
        else      { wait_async_0(); }

        const int*      cb = &s_cols[w][buf][0];
        const _Float16* vb = &s_vals[w][buf][0];

        if (r == CHUNK) {
            // hot path: constant trip count, no guards, 4-way acc rotation
#pragma unroll 8
            for (int jj = 0; jj < CHUNK; jj += 4)
                quad(cb, vb, jj, ((jj >> 2) & 1) * 2);
        } else {
            const int r4 = r & ~3;
            for (int jj = 0; jj < r4; jj += 4)
                quad(cb, vb, jj, ((jj >> 2) & 1) * 2);
            // tail (<= 3 elems): assign alternately to half-waves, mask by value
            for (int e = r4; e < r; ++e) {
                const int   col = cb[e];
                const float v   = (((e - r4) & 1) == half) ? (float)vb[e] : 0.0f;
                acc[0] = fmaf(v, xT[(size_t)col * BATCH + b], acc[0]);
            }
        }
    }

    const float accT = (acc[0] + acc[1]) + (acc[2] + acc[3]);

    // ---- cross-half-wave reduction: sum[n] = accT[n] + accT[n+16] ----
    float sum;
#if __has_builtin(__builtin_amdgcn_wmma_f32_16x16x4_f32)
    {
        // A = ones(16x4); B VGPR0 = accT (whichever two B-rows it maps to, the other
        // B VGPR is zero); C = 0.  D[m,n] = accT[n] + accT[n+16] for every m.
        v2f a;  a.x = 1.0f; a.y = 1.0f;
        v2f bm; bm.x = accT; bm.y = 0.0f;
        v8f cm = {};
        cm = __builtin_amdgcn_wmma_f32_16x16x4_f32(false, a, false, bm,
                                                   (short)0, cm, false, false);
        sum = cm[0];   // identical on all lanes' M-rows
    }
#else
    sum = accT + __shfl_xor(accT, 16, 32);
#endif

    if (lane < BATCH) {
        if constexpr (OUT_F16) {
            ((_Float16*)outp)[(size_t)lane * n_rows + row] = (_Float16)sum;  // y^T fp16
        } else {
            ((float*)outp)[(size_t)row * BATCH + lane] = sum;                // t[row][b] f32
        }
    }
}

extern "C" void kernel_launch(void* const* d_in, const int* in_sizes, int n_in,
                              void* d_out, int out_size, void* d_ws, size_t ws_size,
                              hipStream_t stream) {
    const _Float16* x      = (const _Float16*)d_in[0];
    const int*      a_off  = (const int*)d_in[1];
    const int*      a_cols = (const int*)d_in[2];
    const _Float16* a_vals = (const _Float16*)d_in[3];
    const int*      b_off  = (const int*)d_in[4];
    const int*      b_cols = (const int*)d_in[5];
    const _Float16* b_vals = (const _Float16*)d_in[6];

    const int M = in_sizes[1] - 1;          // rows of A
    const int K = in_sizes[4] - 1;          // rows of B == cols of A
    const int N = in_sizes[0] / BATCH;      // cols of B

    float* xT = (float*)d_ws;               // [N][16] f32
    float* t  = xT + (size_t)N * BATCH;     // [K][16] f32

    {
        int total = N * BATCH;
        hipLaunchKernelGGL(transpose_x_kernel, dim3((total + 255) / 256), dim3(256),
                           0, stream, x, xT, N);
    }
    {
        dim3 grid((K + WAVES_PER_BLOCK - 1) / WAVES_PER_BLOCK);
        hipLaunchKernelGGL((spmm_kernel<false>), grid, dim3(THREADS), 0, stream,
                           b_off, b_cols, b_vals, xT, (void*)t, K);
    }
    {
        dim3 grid((M + WAVES_PER_BLOCK - 1) / WAVES_PER_BLOCK);
        hipLaunchKernelGGL((spmm_kernel<true>), grid, dim3(THREADS), 0, stream,
                           a_off, a_cols, a_vals, t, d_out, M);
    }
}